// ResonanceMapping_25280177504890
// MI455X (gfx1250) — compile-verified
//
#include <hip/hip_runtime.h>
#include <hip/hip_bf16.h>
#include <math.h>

// ---------------- problem constants ----------------
#define BB 2
#define SS 512
#define DD 64
#define SD (SS*DD)        // 32768 per batch
#define S2 (SS*SS)        // 262144 per batch
#define MAXN 0.99999f     // 1 - 1e-5
#define ATMAX 0.9999999f  // 1 - 1e-7

typedef float v2f __attribute__((ext_vector_type(2)));
typedef float v8f __attribute__((ext_vector_type(8)));

#define WMMA_F32(a, b, c) \
    __builtin_amdgcn_wmma_f32_16x16x4_f32(false, (a), false, (b), (short)0, (c), false, false)

__device__ __forceinline__ float artanhf_(float x) {
    x = fminf(fmaxf(x, -ATMAX), ATMAX);
    return 0.5f * logf((1.f + x) / (1.f - x));
}

// hash / PRNG (deterministic surrogate for jax PRNG streams)
__device__ __forceinline__ unsigned pcg(unsigned v) {
    unsigned s = v * 747796405u + 2891336453u;
    unsigned w = ((s >> ((s >> 28) + 4u)) ^ s) * 277803737u;
    return (w >> 22) ^ w;
}
__device__ __forceinline__ float unif01(unsigned v) {
    return ((float)pcg(v) + 0.5f) * (1.0f / 4294967296.0f);
}

// block reductions (power-of-two block)
__device__ __forceinline__ float redSum(float* sm, int t, int n, float v) {
    sm[t] = v; __syncthreads();
    for (int o = n >> 1; o > 0; o >>= 1) { if (t < o) sm[t] += sm[t + o]; __syncthreads(); }
    float r = sm[0]; __syncthreads();
    return r;
}
__device__ __forceinline__ float redMin(float* sm, int t, int n, float v) {
    sm[t] = v; __syncthreads();
    for (int o = n >> 1; o > 0; o >>= 1) { if (t < o) sm[t] = fminf(sm[t], sm[t + o]); __syncthreads(); }
    float r = sm[0]; __syncthreads();
    return r;
}

// =====================================================================
// WMMA GEMM kernels (V_WMMA_F32_16X16X4_F32, one wave per block, EXEC all-1s)
// A fragment: lane l -> A[m0 + l%16][kk + 2*(l/16) + {0,1}]  (b64 load)
// B fragment: lane l -> B[kk + 2*(l/16) + {0,1}][n0 + l%16]
// C/D layout: vgpr j, lane l -> C[m0 + j + 8*(l/16)][n0 + l%16]
// =====================================================================

// C[512x512] = A[512x64] * B[512x64]^T.  16x64 N-strip per wave: A fragment
// reused across 4 accumulators; B^T fragments are contiguous b64 loads.
__global__ void k_gemm_ABt(const float* __restrict__ A, const float* __restrict__ Bm,
                           float* __restrict__ C) {
    int b = blockIdx.z;
    const float* Ab = A + (long)b * SD;
    const float* Bb = Bm + (long)b * SD;
    float* Cb = C + (long)b * S2;
    int n0 = blockIdx.x * 64, m0 = blockIdx.y * 16;
    int lane = threadIdx.x, hi = lane >> 4, lo = lane & 15;
    const float* ap = Ab + (m0 + lo) * DD + 2 * hi;
    const float* bp = Bb + (n0 + lo) * DD + 2 * hi;  // B^T: contiguous in K
    v8f acc[4] = {{}, {}, {}, {}};
#pragma unroll
    for (int kk = 0; kk < DD; kk += 4) {
        v2f av = { ap[kk], ap[kk + 1] };
#pragma unroll
        for (int nt = 0; nt < 4; nt++) {
            v2f bv = { bp[nt * 16 * DD + kk], bp[nt * 16 * DD + kk + 1] };
            acc[nt] = WMMA_F32(av, bv, acc[nt]);
        }
    }
#pragma unroll
    for (int nt = 0; nt < 4; nt++)
#pragma unroll
        for (int j = 0; j < 8; j++)
            Cb[(long)(m0 + j + 8 * hi) * SS + n0 + nt * 16 + lo] = acc[nt][j];
}

// C[512x64] = A[512x512] * B[512x64]  (K=512).  64x16 M-strip per wave:
// the strided B fragment (2 scalar loads/step) is reused across 4 m-tiles.
__global__ void k_gemm_AB64(const float* __restrict__ A, const float* __restrict__ Bm,
                            float* __restrict__ C) {
    int b = blockIdx.z;
    const float* Ab = A + (long)b * S2;
    const float* Bb = Bm + (long)b * SD;
    float* Cb = C + (long)b * SD;
    int n0 = blockIdx.x * 16, m0 = blockIdx.y * 64;
    int lane = threadIdx.x, hi = lane >> 4, lo = lane & 15;
    const float* ap = Ab + (long)(m0 + lo) * SS + 2 * hi;
    v8f acc[4] = {{}, {}, {}, {}};
    for (int kk = 0; kk < SS; kk += 4) {
        int kr = kk + 2 * hi;
        v2f bv = { Bb[kr * DD + n0 + lo], Bb[(kr + 1) * DD + n0 + lo] };
#pragma unroll
        for (int mt = 0; mt < 4; mt++) {
            v2f av = { ap[(long)mt * 16 * SS + kk], ap[(long)mt * 16 * SS + kk + 1] };
            acc[mt] = WMMA_F32(av, bv, acc[mt]);
        }
    }
#pragma unroll
    for (int mt = 0; mt < 4; mt++)
#pragma unroll
        for (int j = 0; j < 8; j++)
            Cb[(long)(m0 + mt * 16 + j + 8 * hi) * DD + n0 + lo] = acc[mt][j];
}

// projections: {q,k,v} = X @ W^T + bias  (grid.z = b*3 + which); 16x64 strip/wave
__global__ void k_proj(const float* __restrict__ qi, const float* __restrict__ ki,
                       const float* __restrict__ vi,
                       const float* __restrict__ Wq, const float* __restrict__ bq,
                       const float* __restrict__ Wk, const float* __restrict__ bk,
                       const float* __restrict__ Wv, const float* __restrict__ bv,
                       float* __restrict__ qp, float* __restrict__ kp, float* __restrict__ vp) {
    int z = blockIdx.z, b = z / 3, t = z % 3;
    const float* X = (t == 0) ? qi : (t == 1) ? ki : vi;
    const float* W = (t == 0) ? Wq : (t == 1) ? Wk : Wv;
    const float* bi = (t == 0) ? bq : (t == 1) ? bk : bv;
    float* O = (t == 0) ? qp : (t == 1) ? kp : vp;
    X += (long)b * SD; O += (long)b * SD;
    int m0 = blockIdx.y * 16;
    int lane = threadIdx.x, hi = lane >> 4, lo = lane & 15;
    const float* ap = X + (m0 + lo) * DD + 2 * hi;
    const float* bp = W + lo * DD + 2 * hi;   // W^T: contiguous in K
    v8f acc[4] = {{}, {}, {}, {}};
#pragma unroll
    for (int kk = 0; kk < DD; kk += 4) {
        v2f av = { ap[kk], ap[kk + 1] };
#pragma unroll
        for (int nt = 0; nt < 4; nt++) {
            v2f bv2 = { bp[nt * 16 * DD + kk], bp[nt * 16 * DD + kk + 1] };
            acc[nt] = WMMA_F32(av, bv2, acc[nt]);
        }
    }
#pragma unroll
    for (int nt = 0; nt < 4; nt++) {
        float bias = bi[nt * 16 + lo];
#pragma unroll
        for (int j = 0; j < 8; j++)
            O[(long)(m0 + j + 8 * hi) * DD + nt * 16 + lo] = acc[nt][j] + bias;
    }
}

// mx = hp_hyp @ o_weight^T  (f64 weights converted on load); 16x64 strip/wave
__global__ void k_gemm_oW(const float* __restrict__ X, const double* __restrict__ OW,
                          float* __restrict__ C) {
    int b = blockIdx.z;
    const float* Xb = X + (long)b * SD;
    float* Cb = C + (long)b * SD;
    int m0 = blockIdx.y * 16;
    int lane = threadIdx.x, hi = lane >> 4, lo = lane & 15;
    const float* ap = Xb + (m0 + lo) * DD + 2 * hi;
    const double* bp = OW + lo * DD + 2 * hi;
    v8f acc[4] = {{}, {}, {}, {}};
#pragma unroll
    for (int kk = 0; kk < DD; kk += 4) {
        v2f av = { ap[kk], ap[kk + 1] };
#pragma unroll
        for (int nt = 0; nt < 4; nt++) {
            v2f bv = { (float)bp[nt * 16 * DD + kk], (float)bp[nt * 16 * DD + kk + 1] };
            acc[nt] = WMMA_F32(av, bv, acc[nt]);
        }
    }
#pragma unroll
    for (int nt = 0; nt < 4; nt++)
#pragma unroll
        for (int j = 0; j < 8; j++)
            Cb[(long)(m0 + j + 8 * hi) * DD + nt * 16 + lo] = acc[nt][j];
}

// =====================================================================
// Elementwise / row kernels
// =====================================================================

// q_hyp = expmap0(qp); k_hyp = expmap0(kp); k_tan = logmap0(k_hyp); |.|^2 scalars
__global__ void k_maps(const float* __restrict__ qp, const float* __restrict__ kp,
                       float* __restrict__ qh, float* __restrict__ kh,
                       float* __restrict__ ktan, float* __restrict__ x2q,
                       float* __restrict__ y2k) {
    __shared__ float sm[DD];
    int i = blockIdx.x, b = blockIdx.y, which = blockIdx.z, t = threadIdx.x;
    long ro = ((long)b * SS + i) * DD;
    float u = (which ? kp : qp)[ro + t];
    float n2 = redSum(sm, t, DD, u * u);
    float nu = sqrtf(n2);
    float n = fmaxf(nu, 1e-15f);
    float th = tanhf(n);
    float y = th * u / n;
    float ny = th * nu / n;
    float pf = (ny > MAXN) ? MAXN / ny : 1.f;
    y *= pf; ny *= pf;
    if (which) {
        kh[ro + t] = y;
        float nn = fmaxf(ny, 1e-15f);
        ktan[ro + t] = artanhf_(ny) * y / nn;
        if (t == 0) y2k[b * SS + i] = ny * ny;
    } else {
        qh[ro + t] = y;
        if (t == 0) x2q[b * SS + i] = ny * ny;
    }
}

// dist row -> rowmin -> w = exp(-(dist-min)); mass = sum(w)
__global__ void k_dist_w(float* __restrict__ scr0, const float* __restrict__ x2q,
                         const float* __restrict__ y2k, float* __restrict__ mass) {
    __shared__ float sm[256];
    int i = blockIdx.x, b = blockIdx.y, t = threadIdx.x;
    long ro = ((long)b * SS + i) * SS;
    float x2 = x2q[b * SS + i];
    float dl[2], mn = 3.4e38f;
#pragma unroll
    for (int jj = 0; jj < 2; jj++) {
        int j = t + jj * 256;
        float xy = scr0[ro + j];
        float y2 = y2k[b * SS + j];
        float d2 = fmaxf(x2 + y2 - 2.f * xy, 0.f);
        float den = fmaxf(1.f - 2.f * xy + x2 * y2, 1e-15f);
        float dist = 2.f * artanhf_(sqrtf(fmaxf(d2 / den, 1e-30f)));
        dl[jj] = dist; mn = fminf(mn, dist);
    }
    mn = redMin(sm, t, 256, mn);
    float sum = 0.f;
#pragma unroll
    for (int jj = 0; jj < 2; jj++) {
        float w = expf(-(dl[jj] - mn));
        scr0[ro + t + jj * 256] = w;
        sum += w;
    }
    sum = redSum(sm, t, 256, sum);
    if (t == 0) mass[b * SS + i] = sum;
}

// cen = clip(expmap0((w@ktan)/(mass+1e-8))); x2c; origin_dist
__global__ void k_cen(float* __restrict__ cen, const float* __restrict__ mass,
                      float* __restrict__ x2c, float* __restrict__ o_odist) {
    __shared__ float sm[DD];
    int i = blockIdx.x, b = blockIdx.y, t = threadIdx.x;
    long ro = ((long)b * SS + i) * DD;
    float pre = cen[ro + t] / (mass[b * SS + i] + 1e-8f);
    float n2 = redSum(sm, t, DD, pre * pre);
    float nu = sqrtf(n2), n = fmaxf(nu, 1e-15f);
    float th = tanhf(n);
    float y = th * pre / n;
    float ny = th * nu / n;
    float pf = (ny > MAXN) ? MAXN / ny : 1.f;
    y *= pf; ny *= pf;
    float f = fminf(MAXN / (ny + 1e-8f), 1.f);   // reference's extra clip
    y *= f; float cn = ny * f;
    cen[ro + t] = y;
    if (t == 0) {
        x2c[b * SS + i] = cn * cn;
        o_odist[b * SS + i] = 2.f * artanhf_(cn);
    }
}

// vr init: hash-normal, row-normalized (surrogate for jax normal key 42)
__global__ void k_vr_init(float* __restrict__ vr) {
    __shared__ float sm[DD];
    int i = blockIdx.x, b = blockIdx.y, t = threadIdx.x;
    long ro = ((long)b * SS + i) * DD;
    unsigned idx = (unsigned)(ro + t);
    float u1 = unif01(idx * 2u + 42u), u2 = unif01(idx * 2u + 1u + 42u);
    float g = sqrtf(-2.f * logf(u1)) * cosf(6.2831853f * u2);
    float n2 = redSum(sm, t, DD, g * g);
    vr[ro + t] = g / fmaxf(sqrtf(n2), 1e-12f);
}

// c2k^2, variance/tension, and rank-2 factor coefficients a,b of sqrt(w)*k_local
__global__ void k_geom(const float* __restrict__ xyck, const float* __restrict__ wbuf,
                       const float* __restrict__ x2c, const float* __restrict__ y2k,
                       const float* __restrict__ mass, float* __restrict__ abuf,
                       float* __restrict__ bbuf, float* __restrict__ o_c2k2,
                       float* __restrict__ o_tension, float* __restrict__ varr,
                       const float* __restrict__ tau) {
    __shared__ float sm[256];
    int i = blockIdx.x, b = blockIdx.y, t = threadIdx.x;
    long ro = ((long)b * SS + i) * SS;
    float x2 = x2c[b * SS + i];
    float var = 0.f;
#pragma unroll
    for (int jj = 0; jj < 2; jj++) {
        int j = t + jj * 256;
        float xy = xyck[ro + j];
        float y2 = y2k[b * SS + j];
        float w = wbuf[ro + j];
        float den = fmaxf(1.f - 2.f * xy + x2 * y2, 1e-15f);
        float d2 = fmaxf(x2 + y2 - 2.f * xy, 0.f);
        float c2k = 2.f * artanhf_(sqrtf(fmaxf(d2 / den, 1e-30f)));
        float c2 = c2k * c2k;
        o_c2k2[ro + j] = c2;
        var += w * c2;
        float c1 = 1.f - 2.f * xy + y2;
        float cc2 = 1.f - x2;
        float m2 = (c1 * c1 * x2 - 2.f * c1 * cc2 * xy + cc2 * cc2 * y2) / (den * den);
        float mnv = fmaxf(sqrtf(fmaxf(m2, 0.f)), 1e-15f);
        float scale = cc2 * artanhf_(mnv) / mnv;       // (1-x2)*artanh(|m|)/|m|
        float sw = sqrtf(w + 1e-8f);
        abuf[ro + j] = -sw * scale * c1 / den;
        bbuf[ro + j] = sw * scale * cc2 / den;
    }
    var = redSum(sm, t, 256, var);
    if (t == 0) {
        varr[b * SS + i] = var;
        o_tension[b * SS + i] = var - tau[0] * (mass[b * SS + i] + 1e-8f);
    }
}

// power-iteration middle: p = a*cv + b*kv; t1 = sum(a*p); scr1 <- g = b*p
__global__ void k_combine(float* __restrict__ scr1, const float* __restrict__ abuf,
                          const float* __restrict__ bbuf, const float* __restrict__ cen,
                          const float* __restrict__ vr, float* __restrict__ t1) {
    __shared__ float sm[256];
    int i = blockIdx.x, b = blockIdx.y, t = threadIdx.x;
    long roS = ((long)b * SS + i) * SS;
    long roD = ((long)b * SS + i) * DD;
    float cvp = (t < DD) ? cen[roD + t] * vr[roD + t] : 0.f;
    float cv = redSum(sm, t, 256, cvp);
    float t1l = 0.f;
#pragma unroll
    for (int jj = 0; jj < 2; jj++) {
        int j = t + jj * 256;
        float a = abuf[roS + j], bb = bbuf[roS + j];
        float kv = scr1[roS + j];
        float p = a * cv + bb * kv;
        t1l += a * p;
        scr1[roS + j] = bb * p;
    }
    t1l = redSum(sm, t, 256, t1l);
    if (t == 0) t1[b * SS + i] = t1l;
}

// vr = normalize(gk + cen * t1)
__global__ void k_vr_upd(const float* __restrict__ gk, const float* __restrict__ cen,
                         const float* __restrict__ t1, float* __restrict__ vr) {
    __shared__ float sm[DD];
    int i = blockIdx.x, b = blockIdx.y, t = threadIdx.x;
    long ro = ((long)b * SS + i) * DD;
    float v = gk[ro + t] + cen[ro + t] * t1[b * SS + i];
    float n2 = redSum(sm, t, DD, v * v);
    vr[ro + t] = v / fmaxf(sqrtf(n2), 1e-12f);
}

// asinh compression, w_proj select, pitchfork bifurcation, expmap0 of message
__global__ void k_mix(const float* __restrict__ hbuf, const float* __restrict__ vr,
                      const float* __restrict__ x2c, const float* __restrict__ x2q,
                      const float* __restrict__ qh, const float* __restrict__ cen,
                      const float* __restrict__ varr, const float* __restrict__ o_tension,
                      float* __restrict__ hcb, float* __restrict__ o_wproj,
                      float* __restrict__ o_x, float* __restrict__ o_hnorm,
                      float* __restrict__ o_hcn, const float* __restrict__ ts) {
    __shared__ float sm[DD];
    int i = blockIdx.x, b = blockIdx.y, t = threadIdx.x;
    long ro = ((long)b * SS + i) * DD;
    int rs = b * SS + i;
    float h = hbuf[ro + t];
    float hn = sqrtf(redSum(sm, t, DD, h * h));
    float hc = asinhf(hn) * h / (hn + 1e-8f);
    float hcn = sqrtf(redSum(sm, t, DD, hc * hc));
    float x2cv = x2c[rs];
    float wpg = vr[ro + t] / fmaxf(1.f - x2cv, 1e-15f);
    float qnrm = sqrtf(x2q[rs]);
    float qt = artanhf_(qnrm) * qh[ro + t] / fmaxf(qnrm, 1e-15f);
    float cnrm = sqrtf(x2cv);
    float ct = artanhf_(cnrm) * cen[ro + t] / fmaxf(cnrm, 1e-15f);
    float diff = qt - ct;
    float dn = sqrtf(redSum(sm, t, DD, diff * diff));
    float fb = diff / fmaxf(dn, 1e-8f);
    float wp = (varr[rs] > 1e-5f) ? wpg : fb;
    o_wproj[ro + t] = wp;
    float x = redSum(sm, t, DD, hc * wp);
    float tension = o_tension[rs];
    float amp = hcn * tanhf(tension) * ((tension > 0.f) ? 1.f : 0.f);
    // deterministic gumbel-argmax surrogate (key 7)
    unsigned base = (unsigned)rs * 2u + 7u * 2654435761u;
    float g0 = -logf(-logf(unif01(base)));
    float g1 = -logf(-logf(unif01(base + 1u)));
    float eps = (g0 > g1) ? 1.f : -1.f;
    float hp = hc + (eps * amp - x) * wp;
    float u = hp * ts[0];
    // expmap0(u)
    float n2 = redSum(sm, t, DD, u * u);
    float nu = sqrtf(n2), n = fmaxf(nu, 1e-15f);
    float th = tanhf(n);
    float y = th * u / n;
    float ny = th * nu / n;
    float pf = (ny > MAXN) ? MAXN / ny : 1.f;
    hcb[ro + t] = y * pf;
    if (t == 0) { o_x[rs] = x; o_hnorm[rs] = hn; o_hcn[rs] = hcn; }
}

// Mobius matvec tail + final logmap0
__global__ void k_final(const float* __restrict__ mxb, const float* __restrict__ hcb,
                        float* __restrict__ o_out) {
    __shared__ float sm[DD];
    int i = blockIdx.x, b = blockIdx.y, t = threadIdx.x;
    long ro = ((long)b * SS + i) * DD;
    float xa = hcb[ro + t];
    float xn = sqrtf(redSum(sm, t, DD, xa * xa));
    float xn_s = fmaxf(xn, 1e-15f);
    float mx = mxb[ro + t];
    float mxn = sqrtf(redSum(sm, t, DD, mx * mx));
    float mxn_s = fmaxf(mxn, 1e-15f);
    float tv = tanhf(mxn_s / xn_s * artanhf_(xn));
    float r = tv * mx / mxn_s;
    if (mxn <= 1e-15f) r = 0.f;
    float rn = sqrtf(redSum(sm, t, DD, r * r));
    float pf = (rn > MAXN) ? MAXN / rn : 1.f;
    r *= pf; rn *= pf;
    o_out[ro + t] = artanhf_(rn) * r / fmaxf(rn, 1e-15f);
}

// =====================================================================
extern "C" void kernel_launch(void* const* d_in, const int* in_sizes, int n_in,
                              void* d_out, int out_size, void* d_ws, size_t ws_size,
                              hipStream_t stream) {
    const float* q_in = (const float*)d_in[0];
    const float* k_in = (const float*)d_in[1];
    const float* v_in = (const float*)d_in[2];
    const float* Wq = (const float*)d_in[3];
    const float* bq = (const float*)d_in[4];
    const float* Wk = (const float*)d_in[5];
    const float* bk = (const float*)d_in[6];
    const float* Wv = (const float*)d_in[7];
    const float* bv = (const float*)d_in[8];
    const double* oW = (const double*)d_in[9];
    const float* ts = (const float*)d_in[10];
    const float* tau = (const float*)d_in[11];

    float* out = (float*)d_out;
    float* o_out = out;                 // [B,S,D]
    float* o_c2k2 = out + 65536;        // [B,S,S]
    float* o_tension = out + 589824;    // [B,S]
    float* o_wproj = out + 590848;      // [B,S,D]
    float* o_odist = out + 656384;      // [B,S]
    float* o_x = out + 657408;          // [B,S]
    float* o_hnorm = out + 658432;      // [B,S]
    float* o_hcn = out + 659456;        // [B,S]

    float* ws = (float*)d_ws;           // needs ~10.8 MB
    float* qp = ws + 0;
    float* kp = ws + 65536;
    float* vp = ws + 131072;
    float* qh = ws + 196608;
    float* kh = ws + 262144;
    float* ktan = ws + 327680;
    float* hbuf = ws + 393216;
    float* cenb = ws + 458752;
    float* vrb = ws + 524288;
    float* hcb = ws + 589824;
    float* mxb = ws + 655360;
    float* scr0 = ws + 720896;          // [B,S,S] : xy_qk -> w
    float* scr1 = ws + 1245184;         // [B,S,S] : xy_ck -> kv -> g
    float* abuf = ws + 1769472;         // [B,S,S]
    float* bbuf = ws + 2293760;         // [B,S,S]
    float* x2q = ws + 2818048;
    float* y2k = ws + 2819072;
    float* massb = ws + 2820096;
    float* x2c = ws + 2821120;
    float* varr = ws + 2822144;
    float* t1b = ws + 2823168;

    dim3 wv(32);
    // 1) linear projections (WMMA, 16x64 strip per wave)
    k_proj<<<dim3(1, SS / 16, BB * 3), wv, 0, stream>>>(q_in, k_in, v_in, Wq, bq, Wk, bk, Wv, bv, qp, kp, vp);
    // 2) expmap0 / logmap0 / norms
    k_maps<<<dim3(SS, BB, 2), DD, 0, stream>>>(qp, kp, qh, kh, ktan, x2q, y2k);
    // 3) xy = q_hyp . k_hyp^T (WMMA) -> dist -> w
    k_gemm_ABt<<<dim3(SS / 64, SS / 16, BB), wv, 0, stream>>>(qh, kh, scr0);
    k_dist_w<<<dim3(SS, BB), 256, 0, stream>>>(scr0, x2q, y2k, massb);
    // 4) h = w @ v ; cen_pre = w @ k_tan (WMMA, 64x16 strip per wave)
    k_gemm_AB64<<<dim3(DD / 16, SS / 64, BB), wv, 0, stream>>>(scr0, vp, hbuf);
    k_gemm_AB64<<<dim3(DD / 16, SS / 64, BB), wv, 0, stream>>>(scr0, ktan, cenb);
    k_cen<<<dim3(SS, BB), DD, 0, stream>>>(cenb, massb, x2c, o_odist);
    k_vr_init<<<dim3(SS, BB), DD, 0, stream>>>(vrb);
    // 5) xy_ck = cen . k_hyp^T (WMMA) -> c2k^2, tension, rank-2 coefficients
    k_gemm_ABt<<<dim3(SS / 64, SS / 16, BB), wv, 0, stream>>>(cenb, kh, scr1);
    k_geom<<<dim3(SS, BB), 256, 0, stream>>>(scr1, scr0, x2c, y2k, massb, abuf, bbuf, o_c2k2, o_tension, varr, tau);
    // 6) matrix-free power iteration (3x), each: two WMMA GEMMs + combine
    for (int r = 0; r < 3; r++) {
        k_gemm_ABt<<<dim3(SS / 64, SS / 16, BB), wv, 0, stream>>>(vrb, kh, scr1);
        k_combine<<<dim3(SS, BB), 256, 0, stream>>>(scr1, abuf, bbuf, cenb, vrb, t1b);
        k_gemm_AB64<<<dim3(DD / 16, SS / 64, BB), wv, 0, stream>>>(scr1, kh, mxb);
        k_vr_upd<<<dim3(SS, BB), DD, 0, stream>>>(mxb, cenb, t1b, vrb);
    }
    // 7) mixing / pitchfork / expmap0 of message
    k_mix<<<dim3(SS, BB), DD, 0, stream>>>(hbuf, vrb, x2c, x2q, qh, cenb, varr, o_tension,
                                           hcb, o_wproj, o_x, o_hnorm, o_hcn, ts);
    // 8) mobius matvec (WMMA, f64 weights) + final logmap0
    k_gemm_oW<<<dim3(1, SS / 16, BB), wv, 0, stream>>>(hcb, oW, mxb);
    k_final<<<dim3(SS, BB), DD, 0, stream>>>(mxb, hcb, o_out);
}